// ProtGramDirectGCN_24343874634033
// MI455X (gfx1250) — compile-verified
//
#include <hip/hip_runtime.h>
#include <math.h>

#define D   128
#define CLS 64

typedef __attribute__((ext_vector_type(2))) float v2f;
typedef __attribute__((ext_vector_type(8))) float v8f;

// ---------------------------------------------------------------------------
// fp32 WMMA 16x16 tile: C(16x16) += A(16x128) * W(128x16), K fixed at 128.
// A lane layout (16x4 fp32 A frag): lane L holds A[M=L%16][K = k0 + 2*(L/16) + {0,1}]
// B lane layout (4x16 fp32 B frag): lane L holds W[K = k0 + 2*(L/16) + r][N = L%16]
// C/D layout: VGPR r -> row (r + 8*(L/16)), col (L%16)
// ---------------------------------------------------------------------------
__device__ __forceinline__ v8f wmma_tile_f32(const float* __restrict__ A,
                                             const float* __restrict__ W,
                                             int row0, int col0, int ldw) {
    const int lane = threadIdx.x & 31;
    const int half = lane >> 4;
    const int l16  = lane & 15;
    v8f c = {};
    const float* arow = A + (size_t)(row0 + l16) * D + 2 * half;
    const float* wcol = W + (size_t)(2 * half) * ldw + col0 + l16;
#pragma unroll 4
    for (int k0 = 0; k0 < D; k0 += 4) {
        v2f a = *(const v2f*)(arow + k0);
        v2f b;
        b.x = wcol[(size_t)k0 * ldw];
        b.y = wcol[(size_t)(k0 + 1) * ldw];
        c = __builtin_amdgcn_wmma_f32_16x16x4_f32(false, a, false, b,
                                                  (short)0, c, false, false);
    }
    return c;
}

// h = x + pe (pe broadcast over rows; D == 128 floats flat)
__global__ __launch_bounds__(256) void k_pe_add(const float* __restrict__ x,
                                                const float* __restrict__ pe,
                                                float* __restrict__ h, int n) {
    int i = blockIdx.x * 256 + threadIdx.x;
    if (i < n) h[i] = x[i] + pe[i & (D - 1)];
}

// out[N x 128] = A[N x 128] @ W[128 x 128]   (one wave per 16x16 tile)
__global__ __launch_bounds__(256) void k_gemm128(const float* __restrict__ A,
                                                 const float* __restrict__ W,
                                                 float* __restrict__ out, int nrows) {
    int wid = (blockIdx.x * 256 + threadIdx.x) >> 5;
    int ntiles = (nrows >> 4) * (D / 16);
    if (wid >= ntiles) return;
    int row0 = (wid >> 3) << 4;
    int col0 = (wid & 7) << 4;
    v8f c = wmma_tile_f32(A, W, row0, col0, D);
    int lane = threadIdx.x & 31, half = lane >> 4, l16 = lane & 15;
    float* o = out + (size_t)(row0 + 8 * half) * D + col0 + l16;
#pragma unroll
    for (int r = 0; r < 8; ++r) o[(size_t)r * D] = c[r];
}

// acc = (Cin+Cout) * (A@Ws) + Cin*(bmi+bsi) + Cout*(bmo+bso)
__global__ __launch_bounds__(256) void k_gemm_sharedacc(
    const float* __restrict__ A,  const float* __restrict__ Ws,
    const float* __restrict__ Ci, const float* __restrict__ Co,
    const float* __restrict__ bmi, const float* __restrict__ bsi,
    const float* __restrict__ bmo, const float* __restrict__ bso,
    float* __restrict__ acc, int nrows) {
    int wid = (blockIdx.x * 256 + threadIdx.x) >> 5;
    int ntiles = (nrows >> 4) * (D / 16);
    if (wid >= ntiles) return;
    int row0 = (wid >> 3) << 4;
    int col0 = (wid & 7) << 4;
    v8f c = wmma_tile_f32(A, Ws, row0, col0, D);
    int lane = threadIdx.x & 31, half = lane >> 4, l16 = lane & 15;
    int col = col0 + l16;
    float bi = bmi[col] + bsi[col];
    float bo = bmo[col] + bso[col];
#pragma unroll
    for (int r = 0; r < 8; ++r) {
        int row = row0 + 8 * half + r;
        float ci = Ci[row], co = Co[row];
        acc[(size_t)row * D + col] = (ci + co) * c[r] + ci * bi + co * bo;
    }
}

// logits[N x 64] = A[N x 128] @ Wd[128 x 64] + bd
__global__ __launch_bounds__(256) void k_gemm_dec(const float* __restrict__ A,
                                                  const float* __restrict__ Wd,
                                                  const float* __restrict__ bd,
                                                  float* __restrict__ out, int nrows) {
    int wid = (blockIdx.x * 256 + threadIdx.x) >> 5;
    int ntiles = (nrows >> 4) * (CLS / 16);
    if (wid >= ntiles) return;
    int row0 = (wid >> 2) << 4;
    int col0 = (wid & 3) << 4;
    v8f c = wmma_tile_f32(A, Wd, row0, col0, CLS);
    int lane = threadIdx.x & 31, half = lane >> 4, l16 = lane & 15;
    int col = col0 + l16;
    float b = bd[col];
    float* o = out + (size_t)(row0 + 8 * half) * CLS + col;
#pragma unroll
    for (int r = 0; r < 8; ++r) o[(size_t)r * CLS] = c[r] + b;
}

// acc[dst] += Cdst[dst] * ew[e] * msg[src]   — one wave per edge, 4 floats/lane
__global__ __launch_bounds__(256) void k_scatter(const float* __restrict__ msg,
                                                 const long long* __restrict__ ei,
                                                 const float* __restrict__ ew,
                                                 const float* __restrict__ Cdst,
                                                 float* __restrict__ acc, int E) {
    long long gid = (long long)blockIdx.x * 256 + threadIdx.x;
    int e = (int)(gid >> 5);
    if (e >= E) return;
    int c = ((int)gid & 31) << 2;
    long long src = ei[e];
    long long dst = ei[(long long)E + e];
    float w = ew[e] * Cdst[dst];
    const float* s = msg + src * D + c;
    float*       d = acc + dst * D + c;
    unsafeAtomicAdd(d + 0, w * s[0]);
    unsafeAtomicAdd(d + 1, w * s[1]);
    unsafeAtomicAdd(d + 2, w * s[2]);
    unsafeAtomicAdd(d + 3, w * s[3]);
}

__global__ __launch_bounds__(256) void k_tanh(float* __restrict__ a, int n) {
    int i = blockIdx.x * 256 + threadIdx.x;
    if (i < n) a[i] = tanhf(a[i]);
}

// in-place log_softmax over rows of 64 (one wave per row, 2 elems/lane)
__global__ __launch_bounds__(256) void k_logsoftmax(float* __restrict__ logits, int nrows) {
    int wid = (blockIdx.x * 256 + threadIdx.x) >> 5;
    int lane = threadIdx.x & 31;
    if (wid >= nrows) return;
    float* row = logits + (size_t)wid * CLS;
    float a = row[lane], b = row[lane + 32];
    float m = fmaxf(a, b);
#pragma unroll
    for (int off = 16; off > 0; off >>= 1) m = fmaxf(m, __shfl_xor(m, off, 32));
    float s = expf(a - m) + expf(b - m);
#pragma unroll
    for (int off = 16; off > 0; off >>= 1) s += __shfl_xor(s, off, 32);
    float lse = m + logf(s);
    row[lane]      = a - lse;
    row[lane + 32] = b - lse;
}

// out = h / max(||h||2, eps) per row of 128 (one wave per row, 4 floats/lane)
__global__ __launch_bounds__(256) void k_normalize(const float* __restrict__ h,
                                                   float* __restrict__ out, int nrows) {
    int wid = (blockIdx.x * 256 + threadIdx.x) >> 5;
    int lane = threadIdx.x & 31;
    if (wid >= nrows) return;
    const float4 v = *(const float4*)(h + (size_t)wid * D + lane * 4);
    float ss = v.x * v.x + v.y * v.y + v.z * v.z + v.w * v.w;
#pragma unroll
    for (int off = 16; off > 0; off >>= 1) ss += __shfl_xor(ss, off, 32);
    float inv = 1.0f / fmaxf(sqrtf(ss), 1e-12f);
    float4 o = make_float4(v.x * inv, v.y * inv, v.z * inv, v.w * inv);
    *(float4*)(out + (size_t)wid * D + lane * 4) = o;
}

extern "C" void kernel_launch(void* const* d_in, const int* in_sizes, int n_in,
                              void* d_out, int out_size, void* d_ws, size_t ws_size,
                              hipStream_t stream) {
    const float*     x      = (const float*)d_in[0];
    const long long* ei_in  = (const long long*)d_in[1];
    const float*     ew_in  = (const float*)d_in[2];
    const long long* ei_out = (const long long*)d_in[3];
    const float*     ew_out = (const float*)d_in[4];
    const float*     pe     = (const float*)d_in[5];
    const float*     Wdec   = (const float*)d_in[24];
    const float*     bdec   = (const float*)d_in[25];

    const int N = in_sizes[0] / D;      // 50000 (multiple of 16)
    const int E = in_sizes[2];          // 800000
    const int elems = N * D;

    float* P0 = (float*)d_ws;           // 4 ping-pong buffers of N*D floats
    float* P1 = P0 + (size_t)elems;
    float* P2 = P1 + (size_t)elems;
    float* P3 = P2 + (size_t)elems;
    float* out_logp = (float*)d_out;                 // N * 64
    float* out_hn   = out_logp + (size_t)N * CLS;    // N * 128

    const int eb       = (elems + 255) / 256;
    const int gblocks  = ((N / 16) * (D / 16) * 32 + 255) / 256;
    const int dblocks  = ((N / 16) * (CLS / 16) * 32 + 255) / 256;
    const int sblocks  = (int)(((long long)E * 32 + 255) / 256);
    const int rwblocks = (N * 32 + 255) / 256;       // wave-per-row kernels

    k_pe_add<<<eb, 256, 0, stream>>>(x, pe, P0, elems);

    float* h = P0;
    for (int li = 0; li < 2; ++li) {
        const int base = 6 + li * 9;
        const float* Wmi = (const float*)d_in[base + 0];
        const float* Wmo = (const float*)d_in[base + 1];
        const float* Ws  = (const float*)d_in[base + 2];
        const float* bmi = (const float*)d_in[base + 3];
        const float* bmo = (const float*)d_in[base + 4];
        const float* bsi = (const float*)d_in[base + 5];
        const float* bso = (const float*)d_in[base + 6];
        const float* Ci  = (const float*)d_in[base + 7];
        const float* Co  = (const float*)d_in[base + 8];

        float* mi  = P1;
        float* mo  = P2;
        float* acc = (li == 0) ? P3 : P0;

        k_gemm128<<<gblocks, 256, 0, stream>>>(h, Wmi, mi, N);
        k_gemm128<<<gblocks, 256, 0, stream>>>(h, Wmo, mo, N);
        k_gemm_sharedacc<<<gblocks, 256, 0, stream>>>(h, Ws, Ci, Co,
                                                      bmi, bsi, bmo, bso, acc, N);
        k_scatter<<<sblocks, 256, 0, stream>>>(mi, ei_in,  ew_in,  Ci, acc, E);
        k_scatter<<<sblocks, 256, 0, stream>>>(mo, ei_out, ew_out, Co, acc, E);
        k_tanh<<<eb, 256, 0, stream>>>(acc, elems);
        h = acc;
    }

    k_gemm_dec<<<dblocks, 256, 0, stream>>>(h, Wdec, bdec, out_logp, N);
    k_logsoftmax<<<rwblocks, 256, 0, stream>>>(out_logp, N);
    k_normalize<<<rwblocks, 256, 0, stream>>>(h, out_hn, N);
}